// KronMatching_86517821213147
// MI455X (gfx1250) — compile-verified
//
#include <hip/hip_runtime.h>

typedef __attribute__((ext_vector_type(2))) float v2f;
typedef __attribute__((ext_vector_type(8))) float v8f;
typedef __attribute__((ext_vector_type(4))) unsigned int u32x4;
typedef __attribute__((ext_vector_type(8))) int i32x8;
typedef __attribute__((ext_vector_type(4))) int i32x4;

#define C_DIM 256
#define HW    2304           // 48*48
#define BM    256            // block tile M (rows, from feat_a)
#define BN    128            // block tile N (cols, from feat_b)
#define KC    32
#define NCHUNK (C_DIM / KC)  // 8
// TDM pads 8 dwords after every 128 dwords -> LDS row strides:
#define ASTRIDE 272          // 256 + 2*8 pads per row
#define BSTRIDE 136          // 128 + 8 pad per row
#define ATILE_FLOATS (KC * ASTRIDE)   // 8704 floats = 34 KB
#define BTILE_FLOATS (KC * BSTRIDE)   // 4352 floats = 17 KB

// TDM 2D tile load: KC rows x tile_w floats (row stride HW floats in global),
// into LDS at byte offset lds_addr, with padding 8 dwords per 128 dwords
// (pad_interval=6, pad_amount=7) to de-conflict LDS banks.
__device__ __forceinline__ void tdm_load_tile(const float* gsrc, unsigned lds_addr,
                                              int tile_w) {
    const unsigned long long ga = (unsigned long long)(uintptr_t)gsrc;
    u32x4 g0;
    g0[0] = 1u;                                    // count=1, user descriptor
    g0[1] = lds_addr;                              // lds_addr (bytes)
    g0[2] = (unsigned)(ga & 0xFFFFFFFFu);          // global_addr[31:0]
    g0[3] = (unsigned)((ga >> 32) & 0x01FFFFFFu)   // global_addr[56:32]
          | (2u << 30);                            // type=2 ("image")
    i32x8 g1;
    g1[0] = (int)((2u << 16)                       // data_size=2 (4 bytes)
          |        (1u << 20)                      // pad_enable
          |        (6u << 22)                      // pad_interval: 128 dwords
          |        (7u << 25));                    // pad_amount:   8 dwords
    g1[1] = (int)((HW & 0xFFFFu) << 16);           // tensor_dim0[15:0]
    g1[2] = (int)(((HW >> 16) & 0xFFFFu)           // tensor_dim0[31:16]
          |        ((C_DIM & 0xFFFFu) << 16));     // tensor_dim1[15:0]
    g1[3] = (int)(((C_DIM >> 16) & 0xFFFFu)        // tensor_dim1[31:16]
          |        ((unsigned)tile_w << 16));      // tile_dim0
    g1[4] = (int)(KC & 0xFFFFu);                   // tile_dim1 = 32, tile_dim2 = 0
    g1[5] = (int)HW;                               // tensor_dim0_stride[31:0]
    g1[6] = 0;
    g1[7] = 0;
    const i32x4 z4 = {0, 0, 0, 0};                 // groups 2/3 unused (2D tensor)
    const i32x8 z8 = {0, 0, 0, 0, 0, 0, 0, 0};
    __builtin_amdgcn_tensor_load_to_lds(g0, g1, z4, z4, z8, /*cpol=*/0);
}

__global__ __launch_bounds__(256)
void kron_wmma_f32(const float* __restrict__ A,
                   const float* __restrict__ B,
                   float* __restrict__ O)
{
    // double-buffered padded tiles, [c][m] layout (TDM copies row-major)
    __shared__ float sA[2][ATILE_FLOATS];   // 68 KB
    __shared__ float sB[2][BTILE_FLOATS];   // 34 KB

    const int n  = blockIdx.z;
    const int m0 = blockIdx.y * BM;         // row block (hw of feat_a)
    const int k0 = blockIdx.x * BN;         // col block (ij of feat_b)

    const float* Ab = A + (size_t)n * C_DIM * HW;   // [C][HW]
    const float* Bb = B + (size_t)n * C_DIM * HW;   // [C][HW]
    float*       Ob = O + (size_t)n * HW * HW;      // [HW][HW]

    const int tid  = threadIdx.x;
    const int wave = tid >> 5;
    const int lane = tid & 31;
    const int half = lane >> 4;             // 0: K={0,1}, 1: K={2,3}
    const int l16  = lane & 15;

    const int wr  = wave >> 1;              // 0..3 (M)
    const int wc  = wave & 1;               // 0..1 (N)
    const int wm0 = wr * 64;                // wave tile: 64 (M) x 64 (N)
    const int wn0 = wc * 64;
    const int apad = (wr >= 2) ? 8 : 0;     // mid-row TDM pad for m >= 128

    // keep the compiler from treating TDM-written LDS as never-stored
    if ((int)blockIdx.x == -1) { sA[0][tid] = 0.f; sB[0][tid] = 0.f; }

    const unsigned ldsA = (unsigned)(uintptr_t)(void*)&sA[0][0];
    const unsigned ldsB = (unsigned)(uintptr_t)(void*)&sB[0][0];

    v8f acc[4][4];
#pragma unroll
    for (int mi = 0; mi < 4; ++mi)
#pragma unroll
        for (int ni = 0; ni < 4; ++ni)
            acc[mi][ni] = (v8f)0.0f;

    // ---- prologue: DMA chunk 0 into buffer 0 ----
    if (wave == 0) {
        tdm_load_tile(Ab + m0, ldsA, BM);
        tdm_load_tile(Bb + k0, ldsB, BN);
        __builtin_amdgcn_s_wait_tensorcnt(0);
    }
    __syncthreads();

    for (int chunk = 0; chunk < NCHUNK; ++chunk) {
        // ---- async: DMA next chunk into the other buffer ----
        const int nxt = chunk + 1;
        if (nxt < NCHUNK && wave == 0) {
            const size_t goff = (size_t)nxt * KC * HW;
            const unsigned ba = (unsigned)(nxt & 1) * (ATILE_FLOATS * 4);
            const unsigned bb = (unsigned)(nxt & 1) * (BTILE_FLOATS * 4);
            tdm_load_tile(Ab + goff + m0, ldsA + ba, BM);
            tdm_load_tile(Bb + goff + k0, ldsB + bb, BN);
        }

        const float* bufA = &sA[chunk & 1][0];
        const float* bufB = &sB[chunk & 1][0];

        // ---- compute: 8 k-steps of 4, 16 WMMAs per k-step ----
#pragma unroll
        for (int c = 0; c < KC; c += 4) {
            const int ck = c + 2 * half;
            v2f af[4], bf[4];
#pragma unroll
            for (int mi = 0; mi < 4; ++mi) {
                const int m = wm0 + mi * 16 + l16 + apad;
                v2f a;
                a.x = bufA[(ck    ) * ASTRIDE + m];
                a.y = bufA[(ck + 1) * ASTRIDE + m];
                af[mi] = a;
            }
#pragma unroll
            for (int ni = 0; ni < 4; ++ni) {
                const int nn = wn0 + ni * 16 + l16;
                v2f b;
                b.x = bufB[(ck    ) * BSTRIDE + nn];
                b.y = bufB[(ck + 1) * BSTRIDE + nn];
                bf[ni] = b;
            }
#pragma unroll
            for (int mi = 0; mi < 4; ++mi)
#pragma unroll
                for (int ni = 0; ni < 4; ++ni)
                    acc[mi][ni] = __builtin_amdgcn_wmma_f32_16x16x4_f32(
                        false, af[mi], false, bf[ni],
                        (short)0, acc[mi][ni], false, false);
        }

        // publish next buffer: issuing wave drains TENSORcnt, then barrier
        if (wave == 0) __builtin_amdgcn_s_wait_tensorcnt(0);
        __syncthreads();
    }

    // ---- epilogue: C/D layout -> global (row = half*8 + r, col = l16) ----
#pragma unroll
    for (int mi = 0; mi < 4; ++mi) {
#pragma unroll
        for (int ni = 0; ni < 4; ++ni) {
            const int colg = k0 + wn0 + ni * 16 + l16;
#pragma unroll
            for (int r = 0; r < 8; ++r) {
                const int rowg = m0 + wm0 + mi * 16 + half * 8 + r;
                Ob[(size_t)rowg * HW + colg] = acc[mi][ni][r];
            }
        }
    }
}

extern "C" void kernel_launch(void* const* d_in, const int* in_sizes, int n_in,
                              void* d_out, int out_size, void* d_ws, size_t ws_size,
                              hipStream_t stream) {
    const float* A = (const float*)d_in[0];   // feat_a: [16][256][48][48]
    const float* B = (const float*)d_in[1];   // feat_b: [16][256][48][48]
    float* O = (float*)d_out;                 // out:    [16][2304][2304]
    dim3 grid(HW / BN, HW / BM, 16);          // 18 x 9 x 16 blocks
    kron_wmma_f32<<<grid, 256, 0, stream>>>(A, B, O);
}